// GCN_69191923138874
// MI455X (gfx1250) — compile-verified
//
#include <hip/hip_runtime.h>
#include <hip/hip_bf16.h>

// ---------------------------------------------------------------------------
// GCN (3x GCNConv + BN + ReLU, then FC) for MI455X / gfx1250.
// Dense transforms use V_WMMA_F32_16X16X4_F32 (fp32 WMMA, exact vs reference).
// Each wave owns one 16-row M-tile and ALL N/16 column tiles, so every A
// fragment (global_load_b64) feeds NT WMMAs; W is staged in LDS pre-swizzled
// into per-lane fragment order (conflict-free ds_load_b64).
// Edge aggregation is edge-parallel with fp32 global atomics (L2-resident).
// ---------------------------------------------------------------------------

typedef float v2f __attribute__((ext_vector_type(2)));
typedef float v8f __attribute__((ext_vector_type(8)));

#define N_NODES 100000
#define N_EDGES 3200000
#define BN_EPS  1e-5f

// ---------------------------------------------------------------------------
// WMMA GEMM: H[M x N] = X[M x K] @ W[K x N].  K multiple of 32, N multiple of 16.
// Block = 256 threads = 8 waves; wave w -> M-tile (blockIdx.x*8 + w).
// LDS holds all of W, swizzled as [K/4][NT][lane][2]:
//   entry(kq, nt, lane, j) = W[4*kq + 2*(lane>=16) + j][16*nt + (lane&15)]
// so a lane's B fragment for (k-step kq, column tile nt) is one float2.
// ---------------------------------------------------------------------------
template <int K, int N>
__global__ __launch_bounds__(256) void gcn_gemm_wmma_f32(
    const float* __restrict__ X, const float* __restrict__ W,
    float* __restrict__ H, int M)
{
    constexpr int NT = N / 16;
    __shared__ float Wlds[K * N];

    const int tid = threadIdx.x;

    // Stage + swizzle W into fragment order (coalesced global reads).
    for (int idx = tid; idx < K * N; idx += 256) {
        const int k  = idx / N;
        const int n  = idx - k * N;
        const int kq = k >> 2;
        const int kr = k & 3;
        const int nt = n >> 4;
        const int ln = ((kr >> 1) << 4) | (n & 15);
        Wlds[((((kq * NT + nt) << 5) + ln) << 1) + (kr & 1)] = W[idx];
    }
    __syncthreads();

    const int wave  = tid >> 5;
    const int lane  = tid & 31;
    const int mtile = blockIdx.x * 8 + wave;
    if (mtile * 16 >= M) return;            // wave-uniform exit, EXEC stays full

    const int m0  = mtile * 16;
    const int rlo = lane & 15;
    const int kh  = (lane >> 4) << 1;       // lanes 16..31 hold K+2,K+3 halves

    const float* __restrict__ xrow = X + (size_t)(m0 + rlo) * K;
    const float* __restrict__ blds = Wlds + (lane << 1);

    v8f acc[NT];
#pragma unroll
    for (int nt = 0; nt < NT; ++nt)
        acc[nt] = (v8f){0.f, 0.f, 0.f, 0.f, 0.f, 0.f, 0.f, 0.f};

    for (int kb = 0; kb < K; kb += 32) {
        __builtin_prefetch(xrow + kb + 64, 0, 0);
        for (int k = kb; k < kb + 32; k += 4) {
            // A fragment: lane holds A[M=rlo][k+kh .. k+kh+1]
            const v2f a = *(const v2f*)(xrow + k + kh);
            const int kq = k >> 2;
#pragma unroll
            for (int nt = 0; nt < NT; ++nt) {
                const v2f b = *(const v2f*)(blds + (size_t)(kq * NT + nt) * 64);
                acc[nt] = __builtin_amdgcn_wmma_f32_16x16x4_f32(
                    false, a, false, b, (short)0, acc[nt], false, false);
            }
        }
    }

    // D layout: VGPR j -> (M = j + 8*(lane>=16), N = lane&15)
    float* __restrict__ hout = H + (size_t)m0 * N + rlo;
    const int rofs = (lane >> 4) * 8;
#pragma unroll
    for (int nt = 0; nt < NT; ++nt) {
#pragma unroll
        for (int j = 0; j < 8; ++j)
            hout[(size_t)(j + rofs) * N + nt * 16] = acc[nt][j];
    }
}

// ---------------------------------------------------------------------------
// Degree (in-degree on dst) and deg^{-1/2} (with +1 self loop)
// ---------------------------------------------------------------------------
__global__ __launch_bounds__(256) void gcn_degree(
    const int* __restrict__ dst, float* __restrict__ deg, int E)
{
    int e = blockIdx.x * 256 + threadIdx.x;
    if (e < E) atomicAdd(&deg[dst[e]], 1.0f);
}

__global__ __launch_bounds__(256) void gcn_invsqrt(
    const float* __restrict__ deg, float* __restrict__ dis, int M)
{
    int n = blockIdx.x * 256 + threadIdx.x;
    if (n < M) dis[n] = rsqrtf(deg[n] + 1.0f);
}

// ---------------------------------------------------------------------------
// Edge scatter: agg[dst] += h[src] * dis[src]*dis[dst]
// One thread per (edge, 4-feature chunk). F multiple of 4.
// ---------------------------------------------------------------------------
__global__ __launch_bounds__(256) void gcn_scatter(
    const int* __restrict__ src, const int* __restrict__ dst,
    const float* __restrict__ dis, const float* __restrict__ h,
    float* __restrict__ agg, int F, int F4, int work)
{
    int idx = blockIdx.x * 256 + threadIdx.x;
    if (idx >= work) return;
    const int e = idx / F4;
    const int c = idx - e * F4;
    const int s = src[e];
    const int d = dst[e];
    const float coef = dis[s] * dis[d];
    const float4 v = *(const float4*)(h + (size_t)s * F + c * 4);
    float* out = agg + (size_t)d * F + c * 4;
    atomicAdd(out + 0, v.x * coef);
    atomicAdd(out + 1, v.y * coef);
    atomicAdd(out + 2, v.z * coef);
    atomicAdd(out + 3, v.w * coef);
}

// ---------------------------------------------------------------------------
// Finalize conv (in place on agg): z = agg + h*dis^2 + bias
// ---------------------------------------------------------------------------
__global__ __launch_bounds__(256) void gcn_selfloop_bias(
    float* __restrict__ agg, const float* __restrict__ h,
    const float* __restrict__ dis, const float* __restrict__ bias,
    int F, int total)
{
    int idx = blockIdx.x * 256 + threadIdx.x;
    if (idx >= total) return;
    const int n = idx / F;
    const int f = idx - n * F;
    const float d = dis[n];
    agg[idx] = agg[idx] + h[idx] * (d * d) + bias[f];
}

// ---------------------------------------------------------------------------
// Per-feature sum / sum-of-squares (block-local LDS reduce + global atomics)
// F in {32, 64, 128}; 256/F rows handled per block iteration.
// ---------------------------------------------------------------------------
__global__ __launch_bounds__(256) void gcn_bn_reduce(
    const float* __restrict__ z, float* __restrict__ gsum,
    float* __restrict__ gsq, int F, int M)
{
    __shared__ float s1[256];
    __shared__ float s2[256];
    const int tid  = threadIdx.x;
    const int rows = 256 / F;
    const int f    = tid % F;
    const int r    = tid / F;

    float sum = 0.f, sq = 0.f;
    for (int n = blockIdx.x * rows + r; n < M; n += gridDim.x * rows) {
        const float v = z[(size_t)n * F + f];
        sum += v;
        sq  += v * v;
    }
    s1[tid] = sum;
    s2[tid] = sq;
    __syncthreads();
    if (tid < F) {
        for (int k = 1; k < rows; ++k) {
            sum += s1[tid + k * F];
            sq  += s2[tid + k * F];
        }
        atomicAdd(&gsum[f], sum);
        atomicAdd(&gsq[f], sq);
    }
}

// scale/shift per feature: y = z*scale + shift, relu later
__global__ __launch_bounds__(128) void gcn_bn_params(
    const float* __restrict__ gsum, const float* __restrict__ gsq,
    const float* __restrict__ gamma, const float* __restrict__ beta,
    float* __restrict__ scale, float* __restrict__ shift, int F, float invM)
{
    int f = threadIdx.x;
    if (f >= F) return;
    const float mu  = gsum[f] * invM;
    const float var = gsq[f] * invM - mu * mu;
    const float sc  = gamma[f] * rsqrtf(var + BN_EPS);
    scale[f] = sc;
    shift[f] = beta[f] - mu * sc;
}

__global__ __launch_bounds__(256) void gcn_bn_relu(
    float* __restrict__ z, const float* __restrict__ scale,
    const float* __restrict__ shift, int F, int total)
{
    int idx = blockIdx.x * 256 + threadIdx.x;
    if (idx >= total) return;
    const int f = idx % F;
    const float y = z[idx] * scale[f] + shift[f];
    z[idx] = y > 0.f ? y : 0.f;
}

// ---------------------------------------------------------------------------
// Final FC: out[n] = dot(y[n, 0:32], Wfc) + bfc
// ---------------------------------------------------------------------------
__global__ __launch_bounds__(256) void gcn_fc32(
    const float* __restrict__ y, const float* __restrict__ Wfc,
    const float* __restrict__ bfc, float* __restrict__ out, int M)
{
    int n = blockIdx.x * 256 + threadIdx.x;
    if (n >= M) return;
    const float* row = y + (size_t)n * 32;
    float acc = bfc[0];
#pragma unroll
    for (int f = 0; f < 32; ++f) acc += row[f] * Wfc[f];
    out[n] = acc;
}

// ---------------------------------------------------------------------------
// Host-side orchestration
// ---------------------------------------------------------------------------
extern "C" void kernel_launch(void* const* d_in, const int* in_sizes, int n_in,
                              void* d_out, int out_size, void* d_ws, size_t ws_size,
                              hipStream_t stream)
{
    const int M = N_NODES;
    const int E = N_EDGES;

    const float* x    = (const float*)d_in[0];
    const int*   eidx = (const int*)d_in[1];
    const int*   src  = eidx;
    const int*   dst  = eidx + E;
    const float* W1  = (const float*)d_in[2];
    const float* b1  = (const float*)d_in[3];
    const float* g1  = (const float*)d_in[4];
    const float* bt1 = (const float*)d_in[5];
    const float* W2  = (const float*)d_in[6];
    const float* b2  = (const float*)d_in[7];
    const float* g2  = (const float*)d_in[8];
    const float* bt2 = (const float*)d_in[9];
    const float* W3  = (const float*)d_in[10];
    const float* b3  = (const float*)d_in[11];
    const float* g3  = (const float*)d_in[12];
    const float* bt3 = (const float*)d_in[13];
    const float* Wfc = (const float*)d_in[14];
    const float* bfc = (const float*)d_in[15];
    float* out = (float*)d_out;

    // Workspace layout (floats)
    float* ws = (float*)d_ws;
    size_t o = 0;
    float* h1  = ws + o; o += (size_t)M * 128;
    float* a1  = ws + o; o += (size_t)M * 128;   // agg -> z -> y (layer1)
    float* h2  = ws + o; o += (size_t)M * 64;
    float* a2  = ws + o; o += (size_t)M * 64;
    float* h3  = ws + o; o += (size_t)M * 32;
    float* a3  = ws + o; o += (size_t)M * 32;
    float* deg = ws + o; o += (size_t)M;
    float* dis = ws + o; o += (size_t)M;
    float* gsum  = ws + o; o += 128;
    float* gsq   = ws + o; o += 128;
    float* scale = ws + o; o += 128;
    float* shift = ws + o; o += 128;

    const int gx_m = (M / 16 + 7) / 8;           // 782 blocks of 8 wave-tiles
    const float invM = 1.0f / (float)M;

    // --- degrees (shared by all layers) ---
    hipMemsetAsync(deg, 0, (size_t)M * sizeof(float), stream);
    gcn_degree<<<(E + 255) / 256, 256, 0, stream>>>(dst, deg, E);
    gcn_invsqrt<<<(M + 255) / 256, 256, 0, stream>>>(deg, dis, M);

    // ======================= Layer 1: 256 -> 128 =======================
    {
        const int F = 128, F4 = F / 4, total = M * F;
        gcn_gemm_wmma_f32<256, 128><<<gx_m, 256, 0, stream>>>(x, W1, h1, M);
        hipMemsetAsync(a1, 0, (size_t)total * sizeof(float), stream);
        const int work = E * F4;
        gcn_scatter<<<(work + 255) / 256, 256, 0, stream>>>(src, dst, dis, h1, a1, F, F4, work);
        gcn_selfloop_bias<<<(total + 255) / 256, 256, 0, stream>>>(a1, h1, dis, b1, F, total);
        hipMemsetAsync(gsum, 0, 2 * 128 * sizeof(float), stream);   // gsum+gsq contiguous
        gcn_bn_reduce<<<256, 256, 0, stream>>>(a1, gsum, gsq, F, M);
        gcn_bn_params<<<1, 128, 0, stream>>>(gsum, gsq, g1, bt1, scale, shift, F, invM);
        gcn_bn_relu<<<(total + 255) / 256, 256, 0, stream>>>(a1, scale, shift, F, total);
    }

    // ======================= Layer 2: 128 -> 64 ========================
    {
        const int F = 64, F4 = F / 4, total = M * F;
        gcn_gemm_wmma_f32<128, 64><<<gx_m, 256, 0, stream>>>(a1, W2, h2, M);
        hipMemsetAsync(a2, 0, (size_t)total * sizeof(float), stream);
        const int work = E * F4;
        gcn_scatter<<<(work + 255) / 256, 256, 0, stream>>>(src, dst, dis, h2, a2, F, F4, work);
        gcn_selfloop_bias<<<(total + 255) / 256, 256, 0, stream>>>(a2, h2, dis, b2, F, total);
        hipMemsetAsync(gsum, 0, 2 * 128 * sizeof(float), stream);
        gcn_bn_reduce<<<256, 256, 0, stream>>>(a2, gsum, gsq, F, M);
        gcn_bn_params<<<1, 128, 0, stream>>>(gsum, gsq, g2, bt2, scale, shift, F, invM);
        gcn_bn_relu<<<(total + 255) / 256, 256, 0, stream>>>(a2, scale, shift, F, total);
    }

    // ======================= Layer 3: 64 -> 32 =========================
    {
        const int F = 32, F4 = F / 4, total = M * F;
        gcn_gemm_wmma_f32<64, 32><<<gx_m, 256, 0, stream>>>(a2, W3, h3, M);
        hipMemsetAsync(a3, 0, (size_t)total * sizeof(float), stream);
        const int work = E * F4;
        gcn_scatter<<<(work + 255) / 256, 256, 0, stream>>>(src, dst, dis, h3, a3, F, F4, work);
        gcn_selfloop_bias<<<(total + 255) / 256, 256, 0, stream>>>(a3, h3, dis, b3, F, total);
        hipMemsetAsync(gsum, 0, 2 * 128 * sizeof(float), stream);
        gcn_bn_reduce<<<256, 256, 0, stream>>>(a3, gsum, gsq, F, M);
        gcn_bn_params<<<1, 128, 0, stream>>>(gsum, gsq, g3, bt3, scale, shift, F, invM);
        gcn_bn_relu<<<(total + 255) / 256, 256, 0, stream>>>(a3, scale, shift, F, total);
    }

    // ============================ FC head ==============================
    gcn_fc32<<<(M + 255) / 256, 256, 0, stream>>>(a3, Wfc, bfc, out, M);
}